// GAT_27676769255941
// MI455X (gfx1250) — compile-verified
//
#include <hip/hip_runtime.h>

// ---------------------------------------------------------------------------
// 2-layer GAT (single head) for gfx1250 / MI455X.
// GEMMs: V_WMMA_F32_16X16X4_F32, weights staged in LDS (one copy per block,
// shared by 4 waves). Edge phase: L2-resident gathers + f32 atomics.
// ---------------------------------------------------------------------------

typedef __attribute__((ext_vector_type(2))) float v2f;
typedef __attribute__((ext_vector_type(8))) float v8f;

namespace gat {
constexpr int N    = 100000;
constexpr int DIN  = 128;
constexpr int H    = 64;      // HID == OUT == 64
constexpr int E    = 1600000;
constexpr int NT16 = N / 16;  // 6250 row tiles, exact
constexpr float NEG_SLOPE = 0.2f;
constexpr int KMAX = 128;
constexpr int KPAD = KMAX + 2;  // LDS row pad: keeps 8B alignment, spreads banks
}

// ---------------------------------------------------------------------------
// C[M x 64] = A[M x K] @ B[K x 64] (+ bias row, optional).
// Block = 128 threads = 4 waves. B is staged transposed into LDS once per
// block; each wave then computes one 16x64 tile with 4 WMMA accumulators.
//
// f32 WMMA 16x16x4 operand layout (ISA 7.12.2):
//   A (16x4 MxK):  lane<16: v0=A[M=lane][k+0], v1=A[M=lane][k+1]
//                  lane>=16: v0=A[M=lane-16][k+2], v1=A[M=lane-16][k+3]
//   B (4x16 KxN):  mirrored with N on the low lane bits.
//   C/D (16x16):   VGPR r, lane l -> row = r + (l>=16 ? 8 : 0), col = l%16.
// ---------------------------------------------------------------------------
__global__ __launch_bounds__(128)
void gemm16_wmma_f32(const float* __restrict__ A, const float* __restrict__ B,
                     const float* __restrict__ bias, float* __restrict__ C,
                     int K) {
  __shared__ float ldsB[gat::H * gat::KPAD];   // B^T, padded rows

  // ---- stage B (K x 64, row-major) into LDS transposed: ldsB[c*KPAD + k]
  for (int i = threadIdx.x; i < K * gat::H; i += 128) {
    const int k = i >> 6;        // K index
    const int c = i & 63;        // column
    ldsB[c * gat::KPAD + k] = B[i];
  }
  __syncthreads();

  const int wid  = threadIdx.x >> 5;           // wave 0..3
  const int lane = threadIdx.x & 31;
  const int tile = blockIdx.x * 4 + wid;
  if (tile >= gat::NT16) return;               // wave-uniform: EXEC stays full

  const int l15  = lane & 15;
  const int half = lane >> 4;                  // 0 or 1
  const int row0 = tile * 16;

  v8f acc[4];
#pragma unroll
  for (int t = 0; t < 4; ++t)
#pragma unroll
    for (int r = 0; r < 8; ++r) acc[t][r] = 0.0f;

  const float* arow = A + (size_t)(row0 + l15) * K + half * 2;
  const float* bcol0 = &ldsB[(size_t)l15 * gat::KPAD];   // + t*16*KPAD per tile

  for (int k = 0; k < K; k += 4) {
    v2f a;
    a.x = arow[k + 0];
    a.y = arow[k + 1];
    const int kb = k + half * 2;               // even -> 8B-aligned LDS pair
#pragma unroll
    for (int t = 0; t < 4; ++t) {
      const float* bp = bcol0 + (size_t)t * 16 * gat::KPAD;
      v2f b;
      b.x = bp[kb + 0];
      b.y = bp[kb + 1];
      acc[t] = __builtin_amdgcn_wmma_f32_16x16x4_f32(
          /*neg_a=*/false, a, /*neg_b=*/false, b,
          /*c_mod=*/(short)0, acc[t], /*reuse_a=*/false, /*reuse_b=*/false);
    }
  }

#pragma unroll
  for (int t = 0; t < 4; ++t) {
    const int col = t * 16 + l15;
    const float bv = bias ? bias[col] : 0.0f;
#pragma unroll
    for (int r = 0; r < 8; ++r) {
      const int row = row0 + r + half * 8;
      C[(size_t)row * gat::H + col] = acc[t][r] + bv;
    }
  }
}

// alpha[i] = dot(Hmat[i, 0:64], a[0:64]) — one thread per node, float4 loads.
__global__ void rowdot64(const float* __restrict__ Hmat,
                         const float* __restrict__ a,
                         float* __restrict__ out, int n) {
  int i = blockIdx.x * blockDim.x + threadIdx.x;
  if (i >= n) return;
  const float4* r  = (const float4*)(Hmat + (size_t)i * gat::H);
  const float4* av = (const float4*)a;
  float s = 0.0f;
#pragma unroll
  for (int j = 0; j < 16; ++j) {
    float4 h4 = r[j], a4 = av[j];
    s += h4.x * a4.x + h4.y * a4.y + h4.z * a4.z + h4.w * a4.w;
  }
  out[i] = s;
}

// Order-preserving uint encoding of float for atomicMax-based segment max.
__device__ __forceinline__ unsigned enc_f32(float f) {
  unsigned u = __float_as_uint(f);
  return (u & 0x80000000u) ? ~u : (u | 0x80000000u);
}
__device__ __forceinline__ float dec_f32(unsigned k) {
  unsigned u = (k & 0x80000000u) ? (k & 0x7fffffffu) : ~k;
  return __uint_as_float(u);
}
#define ENC_NEG_INF 0x007FFFFFu   // enc(-inf)

__global__ void init_softmax(unsigned* __restrict__ mkey,
                             float* __restrict__ denom, int n) {
  int i = blockIdx.x * blockDim.x + threadIdx.x;
  if (i >= n) return;
  mkey[i]  = ENC_NEG_INF;
  denom[i] = 0.0f;
}

// logits[e] = leaky_relu(asrc[src] + adst[dst]); segment max via atomicMax.
__global__ void edge_logits(const int* __restrict__ src,
                            const int* __restrict__ dst,
                            const float* __restrict__ asrc,
                            const float* __restrict__ adst,
                            float* __restrict__ elog,
                            unsigned* __restrict__ mkey, int ne) {
  int e = blockIdx.x * blockDim.x + threadIdx.x;
  if (e >= ne) return;
  const int s = src[e], d = dst[e];
  float l = asrc[s] + adst[d];
  l = (l > 0.0f) ? l : gat::NEG_SLOPE * l;
  elog[e] = l;
  atomicMax(&mkey[d], enc_f32(l));
}

// ex[e] = exp(logits - m[dst]); denom[dst] += ex (in-place overwrite of elog).
__global__ void edge_exp(const int* __restrict__ dst,
                         const unsigned* __restrict__ mkey,
                         float* __restrict__ elog,
                         float* __restrict__ denom, int ne) {
  int e = blockIdx.x * blockDim.x + threadIdx.x;
  if (e >= ne) return;
  const int d = dst[e];
  float m = dec_f32(mkey[d]);
  if (!isfinite(m)) m = 0.0f;      // matches reference's isolated-node guard
  const float ex = expf(elog[e] - m);
  elog[e] = ex;
  atomicAdd(&denom[d], ex);
}

// Fold softmax normalization into a single per-edge coefficient.
__global__ void edge_alpha(const int* __restrict__ dst,
                           const float* __restrict__ denom,
                           float* __restrict__ elog, int ne) {
  int e = blockIdx.x * blockDim.x + threadIdx.x;
  if (e >= ne) return;
  elog[e] = elog[e] / (denom[dst[e]] + 1e-16f);
}

// agg[dst, f..f+1] += alpha[e] * h_src[src, f..f+1] — 32 threads per edge.
__global__ void edge_scatter(const int* __restrict__ src,
                             const int* __restrict__ dst,
                             const float* __restrict__ alpha,
                             const float* __restrict__ hsrc,
                             float* __restrict__ agg, int total) {
  int t = blockIdx.x * blockDim.x + threadIdx.x;
  if (t >= total) return;
  const int e = t >> 5;
  const int f = (t & 31) * 2;
  const int s = src[e], d = dst[e];
  const float a = alpha[e];
  const float2 h2 = *(const float2*)&hsrc[(size_t)s * gat::H + f];
  float* dstp = &agg[(size_t)d * gat::H + f];
  atomicAdd(dstp + 0, h2.x * a);
  atomicAdd(dstp + 1, h2.y * a);
}

// out = agg + bias (+ optional ReLU). agg already contains the residual branch.
__global__ void epilogue(const float* __restrict__ agg,
                         const float* __restrict__ bias,
                         float* __restrict__ out, int total, int do_relu) {
  int t = blockIdx.x * blockDim.x + threadIdx.x;
  if (t >= total) return;
  float v = agg[t] + bias[t & 63];
  if (do_relu) v = fmaxf(v, 0.0f);
  out[t] = v;
}

// ---------------------------------------------------------------------------
// Host-side orchestration
// ---------------------------------------------------------------------------
extern "C" void kernel_launch(void* const* d_in, const int* in_sizes, int n_in,
                              void* d_out, int out_size, void* d_ws, size_t ws_size,
                              hipStream_t stream) {
  using namespace gat;
  (void)in_sizes; (void)n_in; (void)out_size; (void)ws_size;

  const float* x      = (const float*)d_in[0];
  const int*   ei     = (const int*)d_in[1];     // [2, E] row-major
  const int*   src    = ei;
  const int*   dst    = ei + E;
  const float* W1_src = (const float*)d_in[2];
  const float* W1_dst = (const float*)d_in[3];
  const float* a1_src = (const float*)d_in[4];
  const float* a1_dst = (const float*)d_in[5];
  const float* b1     = (const float*)d_in[6];
  const float* Wl1    = (const float*)d_in[7];
  const float* bl1    = (const float*)d_in[8];
  const float* W2_src = (const float*)d_in[9];
  const float* W2_dst = (const float*)d_in[10];
  const float* a2_src = (const float*)d_in[11];
  const float* a2_dst = (const float*)d_in[12];
  const float* b2     = (const float*)d_in[13];
  const float* Wl2    = (const float*)d_in[14];
  const float* bl2    = (const float*)d_in[15];
  float* out = (float*)d_out;

  // Workspace carve-up (floats)
  float* ws = (float*)d_ws;
  size_t o = 0;
  float*    h_src = ws + o; o += (size_t)N * H;
  float*    h_dst = ws + o; o += (size_t)N * H;
  float*    agg   = ws + o; o += (size_t)N * H;
  float*    h1    = ws + o; o += (size_t)N * H;
  float*    asrc  = ws + o; o += N;
  float*    adst  = ws + o; o += N;
  unsigned* mkey  = (unsigned*)(ws + o); o += N;
  float*    denom = ws + o; o += N;
  float*    elog  = ws + o; o += E;

  const dim3 blk256(256);
  const int gN     = (N + 255) / 256;
  const int gE     = (E + 255) / 256;
  const int gNH    = (N * H + 255) / 256;
  const int gScat  = (E * 32 + 255) / 256;
  const int gGemm  = (NT16 + 3) / 4;   // 4 waves (tiles) per block

  // ---------------- Layer 1 ----------------
  gemm16_wmma_f32<<<gGemm, 128, 0, stream>>>(x, W1_src, nullptr, h_src, DIN);
  gemm16_wmma_f32<<<gGemm, 128, 0, stream>>>(x, W1_dst, nullptr, h_dst, DIN);
  gemm16_wmma_f32<<<gGemm, 128, 0, stream>>>(x, Wl1,    bl1,     agg,   DIN); // residual seed
  rowdot64<<<gN, blk256, 0, stream>>>(h_src, a1_src, asrc, N);
  rowdot64<<<gN, blk256, 0, stream>>>(h_dst, a1_dst, adst, N);
  init_softmax<<<gN, blk256, 0, stream>>>(mkey, denom, N);
  edge_logits<<<gE, blk256, 0, stream>>>(src, dst, asrc, adst, elog, mkey, E);
  edge_exp<<<gE, blk256, 0, stream>>>(dst, mkey, elog, denom, E);
  edge_alpha<<<gE, blk256, 0, stream>>>(dst, denom, elog, E);
  edge_scatter<<<gScat, blk256, 0, stream>>>(src, dst, elog, h_src, agg, E * 32);
  epilogue<<<gNH, blk256, 0, stream>>>(agg, b1, h1, N * H, /*relu=*/1);

  // ---------------- Layer 2 ----------------
  gemm16_wmma_f32<<<gGemm, 128, 0, stream>>>(h1, W2_src, nullptr, h_src, H);
  gemm16_wmma_f32<<<gGemm, 128, 0, stream>>>(h1, W2_dst, nullptr, h_dst, H);
  gemm16_wmma_f32<<<gGemm, 128, 0, stream>>>(h1, Wl2,    bl2,     agg,   H); // residual seed
  rowdot64<<<gN, blk256, 0, stream>>>(h_src, a2_src, asrc, N);
  rowdot64<<<gN, blk256, 0, stream>>>(h_dst, a2_dst, adst, N);
  init_softmax<<<gN, blk256, 0, stream>>>(mkey, denom, N);
  edge_logits<<<gE, blk256, 0, stream>>>(src, dst, asrc, adst, elog, mkey, E);
  edge_exp<<<gE, blk256, 0, stream>>>(dst, mkey, elog, denom, E);
  edge_alpha<<<gE, blk256, 0, stream>>>(dst, denom, elog, E);
  edge_scatter<<<gScat, blk256, 0, stream>>>(src, dst, elog, h_src, agg, E * 32);
  epilogue<<<gNH, blk256, 0, stream>>>(agg, b2, out, N * H, /*relu=*/0);
}